// DiscreteDiffusion_19636590478081
// MI455X (gfx1250) — compile-verified
//
#include <hip/hip_runtime.h>
#include <hip/hip_bf16.h>
#include <stdint.h>

// Problem constants (from reference)
#define NUM_CLASSES 512
#define LENGTH      2048
#define BATCH       32
// N = 32*2048*512 = 2^25 ; JAX threefry pairs flat index i with i + N/2
#define HALF_N      16777216u   // 16*2048*512 = 2^24

__device__ __forceinline__ uint32_t rotl32(uint32_t x, uint32_t d) {
  return (x << d) | (x >> (32u - d));
}

// Exact JAX threefry2x32 for key = jax.random.key(42) -> [k0,k1] = [0, 42]
__device__ __forceinline__ void threefry2x32_key42(uint32_t x0, uint32_t x1,
                                                   uint32_t& o0, uint32_t& o1) {
  const uint32_t ks0 = 0u;
  const uint32_t ks1 = 42u;
  const uint32_t ks2 = 0x1BD11BDAu ^ 42u;
  x0 += ks0; x1 += ks1;
#define TF_ROUND(r) { x0 += x1; x1 = rotl32(x1, (r)); x1 ^= x0; }
  TF_ROUND(13) TF_ROUND(15) TF_ROUND(26) TF_ROUND(6)
  x0 += ks1; x1 += ks2 + 1u;
  TF_ROUND(17) TF_ROUND(29) TF_ROUND(16) TF_ROUND(24)
  x0 += ks2; x1 += ks0 + 2u;
  TF_ROUND(13) TF_ROUND(15) TF_ROUND(26) TF_ROUND(6)
  x0 += ks0; x1 += ks1 + 3u;
  TF_ROUND(17) TF_ROUND(29) TF_ROUND(16) TF_ROUND(24)
  x0 += ks1; x1 += ks2 + 4u;
  TF_ROUND(13) TF_ROUND(15) TF_ROUND(26) TF_ROUND(6)
  x0 += ks2; x1 += ks0 + 5u;
#undef TF_ROUND
  o0 = x0; o1 = x1;
}

// JAX uniform: u01 = bitcast(bits>>9 | 0x3f800000) - 1 ; u = max(u01, tiny)
// Gumbel g = -log(-log u). We return w = -log(u) > 0; since
// argmax(log p + g) == argmax(p / w) (monotone transform), the caller
// compares p * rcp(w) instead — one transcendental per class, not three.
__device__ __forceinline__ float neglog_u_from_bits(uint32_t bits) {
  float u = __uint_as_float((bits >> 9) | 0x3f800000u) - 1.0f;
  u = fmaxf(u, 1.17549435e-38f);
  return -logf(u);   // in (1.19e-7, 88.0], never 0/inf/nan
}

__global__ __launch_bounds__(256) void dd_sample_kernel(
    const int* __restrict__ x0, const int* __restrict__ t,
    const float* __restrict__ qt, float* __restrict__ out) {
  // 8 waves x 2 rows x 512 floats = 32 KiB LDS
  __shared__ float rows[8 * 2 * NUM_CLASSES];

  const int lane = threadIdx.x & 31;
  const int wave = threadIdx.x >> 5;
  const int b    = blockIdx.y;               // 0..15, paired with b+16
  const int l    = blockIdx.x * 8 + wave;    // 0..2047

  const int xa = x0[b * LENGTH + l];
  const int xb = x0[(b + 16) * LENGTH + l];
  const int ta = t[b];
  const int tb = t[b + 16];

  const float* rowAp = qt + ((size_t)ta * NUM_CLASSES + xa) * NUM_CLASSES;
  const float* rowBp = qt + ((size_t)tb * NUM_CLASSES + xb) * NUM_CLASSES;

  // CDNA5 async global->LDS staging of the two gathered 2 KiB rows.
  // Each wave stages only its own region and waits on its own ASYNCcnt.
  const uint32_t ldsBase =
      (uint32_t)(uintptr_t)(&rows[0]) + (uint32_t)(wave * 2 * NUM_CLASSES * 4);
#pragma unroll
  for (int k = 0; k < 4; ++k) {
    const uint32_t off = (uint32_t)(k * 512 + lane * 16);
    const uint32_t la = ldsBase + off;
    asm volatile("global_load_async_to_lds_b128 %0, %1, %2"
                 :: "v"(la), "v"(off), "s"(rowAp) : "memory");
    const uint32_t lb = ldsBase + 2048u + off;
    asm volatile("global_load_async_to_lds_b128 %0, %1, %2"
                 :: "v"(lb), "v"(off), "s"(rowBp) : "memory");
  }

  // Phase 1 (overlapped with the async gather): the full threefry/Gumbel
  // stream for both rows — depends only on indices, not on the row data.
  const uint32_t base = ((uint32_t)(b * LENGTH + l)) << 9;
  float wA[16], wB[16];
#pragma unroll
  for (int k = 0; k < 16; ++k) {
    const uint32_t c = (uint32_t)(k * 32 + lane);
    uint32_t w0, w1;
    threefry2x32_key42(base + c, base + c + HALF_N, w0, w1);
    wA[k] = neglog_u_from_bits(w0);
    wB[k] = neglog_u_from_bits(w1);
  }

  // Phase 2: rows have long since landed in LDS; score and reduce.
  asm volatile("s_wait_asynccnt 0" ::: "memory");

  const float* rA = &rows[wave * 2 * NUM_CLASSES];
  const float* rB = rA + NUM_CLASSES;

  float bestA = -__builtin_inff(), bestB = -__builtin_inff();
  int idxA = 0, idxB = 0;
#pragma unroll
  for (int k = 0; k < 16; ++k) {
    const int c = k * 32 + lane;   // stride-32: LDS bank-conflict-free
    const float sA = rA[c] * __builtin_amdgcn_rcpf(wA[k]);
    const float sB = rB[c] * __builtin_amdgcn_rcpf(wB[k]);
    if (sA > bestA) { bestA = sA; idxA = c; }   // strict > keeps lowest c in-lane
    if (sB > bestB) { bestB = sB; idxB = c; }
  }

  // wave32 argmax reduction, ties -> lower class index (JAX argmax semantics)
#pragma unroll
  for (int off = 16; off > 0; off >>= 1) {
    const float oA = __shfl_xor(bestA, off);
    const int   jA = __shfl_xor(idxA, off);
    if (oA > bestA || (oA == bestA && jA < idxA)) { bestA = oA; idxA = jA; }
    const float oB = __shfl_xor(bestB, off);
    const int   jB = __shfl_xor(idxB, off);
    if (oB > bestB || (oB == bestB && jB < idxB)) { bestB = oB; idxB = jB; }
  }

  if (lane == 0) {
    out[b * LENGTH + l]        = (float)idxA;   // integer output stored as f32
    out[(b + 16) * LENGTH + l] = (float)idxB;
  }
}

extern "C" void kernel_launch(void* const* d_in, const int* in_sizes, int n_in,
                              void* d_out, int out_size, void* d_ws, size_t ws_size,
                              hipStream_t stream) {
  const int*   x0 = (const int*)d_in[0];    // (32, 2048) int32
  const int*   t  = (const int*)d_in[1];    // (32,) int32
  const float* qt = (const float*)d_in[2];  // (1000, 512, 512) f32
  float* out = (float*)d_out;               // (32, 2048) indices as f32

  dim3 grid(LENGTH / 8, 16);
  dd_sample_kernel<<<grid, 256, 0, stream>>>(x0, t, qt, out);
}